// PointClickLoss_3229815407132
// MI455X (gfx1250) — compile-verified
//
#include <hip/hip_runtime.h>
#include <hip/hip_bf16.h>

typedef float v2f __attribute__((ext_vector_type(2)));
typedef float v8f __attribute__((ext_vector_type(8)));

#define BATCH 32
#define IMG_H 1024
#define IMG_W 1024
// 16 positive + 16 negative points per batch -> one wave32 per batch.

__device__ __forceinline__ float softplus_stable(float z) {
    // jax.nn.softplus(z) = log1p(exp(z)), stable form:
    return fmaxf(z, 0.0f) + log1pf(expf(-fabsf(z)));
}

__global__ __launch_bounds__(1024, 1)
void point_click_loss_kernel(const float* __restrict__ mask,
                             const float* __restrict__ pos_pts,
                             const float* __restrict__ neg_pts,
                             float* __restrict__ out) {
    const int tid  = threadIdx.x;      // 0..1023
    const int b    = tid >> 5;         // batch index  (== wave id)
    const int p    = tid & 31;         // point slot   (== lane id)
    const bool is_pos = (p < 16);

    // ---- gather point (x, y) ----
    const float* pt = is_pos ? (pos_pts + ((size_t)b * 16 + p) * 2)
                             : (neg_pts + ((size_t)b * 16 + (p - 16)) * 2);
    const float x = pt[0];
    const float y = pt[1];

    // ---- bilinear taps (align_corners=True: direct pixel coords) ----
    const float x0f = floorf(x), y0f = floorf(y);
    const float wx1 = x - x0f,   wx0 = 1.0f - wx1;
    const float wy1 = y - y0f,   wy0 = 1.0f - wy1;

    const int x0 = min(max((int)x0f,     0), IMG_W - 1);
    const int x1 = min(max((int)x0f + 1, 0), IMG_W - 1);
    const int y0 = min(max((int)y0f,     0), IMG_H - 1);
    const int y1 = min(max((int)y0f + 1, 0), IMG_H - 1);

    const float* img = mask + (size_t)b * IMG_H * IMG_W;
    const float v00 = img[(size_t)y0 * IMG_W + x0];
    const float v01 = img[(size_t)y0 * IMG_W + x1];
    const float v10 = img[(size_t)y1 * IMG_W + x0];
    const float v11 = img[(size_t)y1 * IMG_W + x1];

    const float logit = wy0 * (wx0 * v00 + wx1 * v01) +
                        wy1 * (wx0 * v10 + wx1 * v11);

    // positive points: softplus(-logit); negative points: softplus(+logit)
    const float val = softplus_stable(is_pos ? -logit : logit);

    // ---- wave reduction: 32 lanes -> 2 partials via exact f32 WMMA ----
    __shared__ float partials[2 * BATCH];
    const int lane = p;

#if __has_builtin(__builtin_amdgcn_wmma_f32_16x16x4_f32)
    // A (16x4 f32): VGPR0 = val -> A[m][0]=v[m] (lanes 0-15), A[m][2]=v[m+16]
    //               VGPR1 = 0  -> A[m][1]=A[m][3]=0
    // B (4x16)    : all ones
    // D[m][n] = v[m] + v[m+16]  (independent of n)
    v2f a;  a[0] = val;  a[1] = 0.0f;
    v2f bb; bb[0] = 1.0f; bb[1] = 1.0f;
    v8f c = {};
    c = __builtin_amdgcn_wmma_f32_16x16x4_f32(
            /*neg_a=*/false, a, /*neg_b=*/false, bb,
            /*c_mod=*/(short)0, c, /*reuse_a=*/false, /*reuse_b=*/false);
    // Per-lane sum over the 8 D VGPRs:
    //   lanes 0-15  : sum_{m=0..7}  (v[m]+v[m+16])
    //   lanes 16-31 : sum_{m=8..15} (v[m]+v[m+16])
    const float wsum = c[0] + c[1] + c[2] + c[3] + c[4] + c[5] + c[6] + c[7];
    if (lane == 0)  partials[2 * b + 0] = wsum;
    if (lane == 16) partials[2 * b + 1] = wsum;
#else
    float wsum = val;
    for (int off = 16; off > 0; off >>= 1)
        wsum += __shfl_xor(wsum, off, 32);
    if (lane == 0) { partials[2 * b + 0] = wsum; partials[2 * b + 1] = 0.0f; }
#endif

    __syncthreads();

    // ---- final fold: 64 partials -> scalar (fixed order, deterministic) ----
    if (tid == 0) {
        float s = 0.0f;
        #pragma unroll
        for (int i = 0; i < 2 * BATCH; ++i) s += partials[i];
        // total = grand_sum / (16 pts * 2 groups * 32 batches) = /1024
        out[0] = s * (1.0f / 1024.0f);
    }
}

extern "C" void kernel_launch(void* const* d_in, const int* in_sizes, int n_in,
                              void* d_out, int out_size, void* d_ws, size_t ws_size,
                              hipStream_t stream) {
    const float* mask = (const float*)d_in[0];   // (32,1,1024,1024) f32
    const float* pos  = (const float*)d_in[1];   // (32,16,2) f32
    const float* neg  = (const float*)d_in[2];   // (32,16,2) f32
    float* out        = (float*)d_out;           // scalar f32

    (void)in_sizes; (void)n_in; (void)out_size; (void)d_ws; (void)ws_size;

    point_click_loss_kernel<<<dim3(1), dim3(1024), 0, stream>>>(mask, pos, neg, out);
}